// MultiHeadDGF_38087769981235
// MI455X (gfx1250) — compile-verified
//
#include <hip/hip_runtime.h>
#include <math.h>

// MI455X / gfx1250, wave32. f32 WMMA 16x16x4 for all GEMM stages; hardware
// v_tanh_f32 (TRANS) + async global->LDS (B128, ASYNCcnt) for the staging.

typedef __attribute__((ext_vector_type(2))) float v2f;
typedef __attribute__((ext_vector_type(8))) float v8f;
typedef __attribute__((ext_vector_type(4))) int v4i;

#define EPSC 1e-6f

#if __has_builtin(__builtin_amdgcn_global_load_async_to_lds_b128) && \
    __has_builtin(__builtin_amdgcn_s_wait_asynccnt)
#define HAVE_ASYNC_LDS 1
#endif

__device__ __forceinline__ v8f wmma_f32_16x16x4(v2f a, v2f b, v8f c) {
  return __builtin_amdgcn_wmma_f32_16x16x4_f32(false, a, false, b, (short)0, c,
                                               false, false);
}

__device__ __forceinline__ float fast_tanh(float x) {
#if __has_builtin(__builtin_amdgcn_tanhf)
  return __builtin_amdgcn_tanhf(x);  // v_tanh_f32 (TRANS, co-executes w/ VALU)
#else
  return tanhf(x);
#endif
}

#ifdef HAVE_ASYNC_LDS
__device__ __forceinline__ void async_cp16(const float* g, void* l) {
  // signature (probe-derived): (v4i* gsrc, v4i* ldsdst, imm offset, imm cpol)
  __builtin_amdgcn_global_load_async_to_lds_b128((v4i*)g, (v4i*)l, 0, 0);
}
#endif

// ---------------------------------------------------------------------------
// Kernel 1: ai[b][h][i][k] = b1[h][k] + x[b,i,:]·W1[h,:128,k]
//           aj[b][h][i][k] =            x[b,i,:]·W1[h,128:,k]
//           sq[b][i]       = ||x[b,i,:]||^2
// ---------------------------------------------------------------------------
__global__ __launch_bounds__(128) void dgf_proj_kernel(
    const float* __restrict__ x, const float* __restrict__ W1,
    const float* __restrict__ b1, float* __restrict__ ai,
    float* __restrict__ aj, float* __restrict__ sq) {
  const int bi = blockIdx.x;  // b*512 + i
  const int t = threadIdx.x;  // 0..127
  const int h = t >> 5, k = t & 31;

  __shared__ float xr[128];
  xr[t] = x[(size_t)bi * 128 + t];
  __syncthreads();

  float a1 = b1[h * 32 + k];
  float a2 = 0.0f;
  const float* w1h = W1 + (size_t)h * 256 * 32 + k;  // W1: [H][2d][hid]
#pragma unroll 4
  for (int dd = 0; dd < 128; ++dd) {
    const float xv = xr[dd];
    a1 = fmaf(xv, w1h[(size_t)dd * 32], a1);
    a2 = fmaf(xv, w1h[(size_t)(128 + dd) * 32], a2);
  }
  const int b = bi >> 9, i = bi & 511;
  const size_t o = (((size_t)b * 4 + h) * 512 + i) * 32 + k;
  ai[o] = a1;
  aj[o] = a2;

  if (t == 0) {
    float s = 0.0f;
    for (int dd = 0; dd < 128; ++dd) s = fmaf(xr[dd], xr[dd], s);
    sq[bi] = s;
  }
}

// ---------------------------------------------------------------------------
// Kernel 2: 128 thr = 4 waves; i-tile 16 rows x 64 j cols (16 per wave).
// Gram tile via f32 WMMA (K=128), then per-lane head/hid tanh MLP ->
// sigma -> adj averaged over heads.
// LDS strides: 132/36 words (%32==4 -> conflict-free; x4 %16==0 -> B128 ok).
// ---------------------------------------------------------------------------
__global__ __launch_bounds__(128) void dgf_pair_kernel(
    const float* __restrict__ x, const float* __restrict__ W2,
    const float* __restrict__ b2, const float* __restrict__ ai,
    const float* __restrict__ aj, const float* __restrict__ sq,
    float* __restrict__ adj) {
  const int jg = blockIdx.x;  // 0..7
  const int it = blockIdx.y;  // 0..31
  const int b = blockIdx.z;   // 0..3
  const int i0 = it * 16, jbase = jg * 64;
  const int t = threadIdx.x;
  const int wave = t >> 5, lane = t & 31;
  const int hf = lane >> 4, r = lane & 15;

  __shared__ float xi_s[16][132];
  __shared__ float ai_s[4][16][36];
  __shared__ float aj_s[4][64][36];
  __shared__ float W2_s[4][32];
  __shared__ float b2_s[4];
  __shared__ float sqi_s[16];
  __shared__ float sqj_s[64];

#ifdef HAVE_ASYNC_LDS
  // xi: 512 x 16B chunks, ai: 512 chunks, aj: 2048 chunks
  for (int idx = t; idx < 16 * 32; idx += 128) {
    const int rr = idx >> 5, cc = (idx & 31) * 4;
    async_cp16(x + ((size_t)b * 512 + i0 + rr) * 128 + cc, &xi_s[rr][cc]);
  }
  for (int idx = t; idx < 4 * 16 * 8; idx += 128) {
    const int h = idx >> 7, rem = idx & 127, m = rem >> 3, kc = (rem & 7) * 4;
    async_cp16(ai + (((size_t)b * 4 + h) * 512 + i0 + m) * 32 + kc,
               &ai_s[h][m][kc]);
  }
  for (int idx = t; idx < 4 * 64 * 8; idx += 128) {
    const int h = idx >> 9, rem = idx & 511, jj = rem >> 3, kc = (rem & 7) * 4;
    async_cp16(aj + (((size_t)b * 4 + h) * 512 + jbase + jj) * 32 + kc,
               &aj_s[h][jj][kc]);
  }
#else
  for (int idx = t; idx < 16 * 128; idx += 128) {
    const int rr = idx >> 7, cc = idx & 127;
    xi_s[rr][cc] = x[((size_t)b * 512 + i0 + rr) * 128 + cc];
  }
  for (int idx = t; idx < 4 * 16 * 32; idx += 128) {
    const int h = idx >> 9, rem = idx & 511, m = rem >> 5, k = rem & 31;
    ai_s[h][m][k] = ai[(((size_t)b * 4 + h) * 512 + i0 + m) * 32 + k];
  }
  for (int idx = t; idx < 4 * 64 * 32; idx += 128) {
    const int h = idx >> 11, rem = idx & 2047, jj = rem >> 5, k = rem & 31;
    aj_s[h][jj][k] = aj[(((size_t)b * 4 + h) * 512 + jbase + jj) * 32 + k];
  }
#endif
  W2_s[t >> 5][t & 31] = W2[t];
  if (t < 4) b2_s[t] = b2[t];
  if (t < 16) sqi_s[t] = sq[b * 512 + i0 + t];
  if (t < 64) sqj_s[t] = sq[b * 512 + jbase + t];
#ifdef HAVE_ASYNC_LDS
  __builtin_amdgcn_s_wait_asynccnt(0);
#endif
  __syncthreads();

  // --- gram tile via WMMA: A = x[i0:,:], B = x[j0:,:]^T, K=128 ---
  const int j0 = jbase + wave * 16;
  const float* xjrow = x + ((size_t)b * 512 + j0 + r) * 128;
  v8f c = {};
  for (int k0 = 0; k0 < 128; k0 += 4) {
    v2f a, bb;
    a.x = xi_s[r][k0 + 2 * hf];
    a.y = xi_s[r][k0 + 2 * hf + 1];
    bb.x = xjrow[k0 + 2 * hf];
    bb.y = xjrow[k0 + 2 * hf + 1];
    c = wmma_f32_16x16x4(a, bb, c);
  }
  // c[v] = gram[i0 + v + 8*hf][j0 + r]

  const int jj = wave * 16 + r;
  const float sqj = sqj_s[jj];
  float d2v[8], acc[8];
#pragma unroll
  for (int v = 0; v < 8; ++v) {
    d2v[v] = fmaxf(sqi_s[v + 8 * hf] + sqj - 2.0f * c[v], 0.0f);
    acc[v] = 0.0f;
  }

  for (int h = 0; h < 4; ++h) {
    float sacc[8];
#pragma unroll
    for (int v = 0; v < 8; ++v) sacc[v] = b2_s[h];
    const float* ajrow = &aj_s[h][jj][0];
    const float* airow = &ai_s[h][8 * hf][0];  // rows v, stride 36 words
#pragma unroll 4
    for (int k = 0; k < 32; ++k) {
      const float ajv = ajrow[k];
      const float w2 = W2_s[h][k];
#pragma unroll
      for (int v = 0; v < 8; ++v) {
        const float pre = fast_tanh(airow[v * 36 + k] + ajv);
        sacc[v] = fmaf(w2, pre, sacc[v]);
      }
    }
#pragma unroll
    for (int v = 0; v < 8; ++v) {
      const float s = sacc[v];
      const float sp = fmaxf(s, 0.0f) + log1pf(__expf(-fabsf(s)));  // softplus
      const float denom = 2.0f * sp * sp + EPSC;
      acc[v] += __expf(-d2v[v] / denom);
    }
  }
#pragma unroll
  for (int v = 0; v < 8; ++v) {
    adj[((size_t)b * 512 + i0 + v + 8 * hf) * 512 + j0 + r] = acc[v] * 0.25f;
  }
}

// ---------------------------------------------------------------------------
// Kernel 3: h = adj_mean @ x (K=512, WMMA) -> LDS -> out = h @ Wp + bp
// (K=128, WMMA). 256 thr = 8 waves per (b, 16-row i-tile).
// ---------------------------------------------------------------------------
__global__ __launch_bounds__(256) void dgf_out_kernel(
    const float* __restrict__ x, const float* __restrict__ Wp,
    const float* __restrict__ bp, const float* __restrict__ adj,
    float* __restrict__ out) {
  const int it = blockIdx.x;  // 0..31
  const int b = blockIdx.y;   // 0..3
  const int i0 = it * 16;
  const int t = threadIdx.x;
  const int wave = t >> 5, lane = t & 31;
  const int hf = lane >> 4, r = lane & 15;
  const int col0 = wave * 16;

  __shared__ float adj_s[16][516];  // 516%64==4 conflict-free; x4%16==0
  __shared__ float h_s[16][132];

#ifdef HAVE_ASYNC_LDS
  for (int idx = t; idx < 16 * 128; idx += 256) {  // 2048 x 16B chunks
    const int rr = idx >> 7, cc = (idx & 127) * 4;
    async_cp16(adj + ((size_t)b * 512 + i0 + rr) * 512 + cc, &adj_s[rr][cc]);
  }
  __builtin_amdgcn_s_wait_asynccnt(0);
#else
  for (int idx = t; idx < 16 * 512; idx += 256) {
    const int rr = idx >> 9, cc = idx & 511;
    adj_s[rr][cc] = adj[((size_t)b * 512 + i0 + rr) * 512 + cc];
  }
#endif
  __syncthreads();

  // h[16x128] = adj_tile[16x512] @ x[512x128]
  v8f c = {};
  for (int k0 = 0; k0 < 512; k0 += 4) {
    v2f a, bb;
    a.x = adj_s[r][k0 + 2 * hf];
    a.y = adj_s[r][k0 + 2 * hf + 1];
    const float* xk = x + ((size_t)b * 512 + k0 + 2 * hf) * 128 + col0 + r;
    bb.x = xk[0];
    bb.y = xk[128];
    c = wmma_f32_16x16x4(a, bb, c);
  }
#pragma unroll
  for (int v = 0; v < 8; ++v) h_s[v + 8 * hf][col0 + r] = c[v];
  __syncthreads();

  // out[16x128] = h[16x128] @ Wp[128x128] + bp
  v8f c2 = {};
  for (int k0 = 0; k0 < 128; k0 += 4) {
    v2f a, bb;
    a.x = h_s[r][k0 + 2 * hf];
    a.y = h_s[r][k0 + 2 * hf + 1];
    const float* wk = Wp + (size_t)(k0 + 2 * hf) * 128 + col0 + r;
    bb.x = wk[0];
    bb.y = wk[128];
    c2 = wmma_f32_16x16x4(a, bb, c2);
  }
  const float bpv = bp[col0 + r];
#pragma unroll
  for (int v = 0; v < 8; ++v) {
    out[((size_t)b * 512 + i0 + v + 8 * hf) * 128 + col0 + r] = c2[v] + bpv;
  }
}

// ---------------------------------------------------------------------------
extern "C" void kernel_launch(void* const* d_in, const int* in_sizes, int n_in,
                              void* d_out, int out_size, void* d_ws,
                              size_t ws_size, hipStream_t stream) {
  const float* x = (const float*)d_in[0];   // [4,512,128]
  const float* W1 = (const float*)d_in[1];  // [4,256,32]
  const float* b1 = (const float*)d_in[2];  // [4,32]
  const float* W2 = (const float*)d_in[3];  // [4,32]
  const float* b2 = (const float*)d_in[4];  // [4]
  const float* Wp = (const float*)d_in[5];  // [128,128]
  const float* bp = (const float*)d_in[6];  // [128]
  float* out = (float*)d_out;               // [4,512,128]

  char* ws = (char*)d_ws;
  float* ai = (float*)(ws);                       // 1 MB
  float* aj = (float*)(ws + (size_t)(1u << 20));  // 1 MB
  float* sq = (float*)(ws + (size_t)(2u << 20));  // 8 KB
  float* adj = (float*)(ws + (size_t)(2u << 20) + (size_t)(64u << 10));  // 4 MB

  dgf_proj_kernel<<<dim3(2048), dim3(128), 0, stream>>>(x, W1, b1, ai, aj, sq);
  dgf_pair_kernel<<<dim3(8, 32, 4), dim3(128), 0, stream>>>(x, W2, b2, ai, aj,
                                                            sq, adj);
  dgf_out_kernel<<<dim3(32, 4), dim3(256), 0, stream>>>(x, Wp, bp, adj, out);
}